// ControlFlowExpert_62380105007397
// MI455X (gfx1250) — compile-verified
//
#include <hip/hip_runtime.h>
#include <hip/hip_bf16.h>

// ---------------------------------------------------------------------------
// Layout constants from the reference
// ---------------------------------------------------------------------------
#define ROW_F        208              // floats per row
#define AX_START     163
#define PC_START     171
#define IMM_START    195
#define BRANCH_TAKEN 203
#define COL_JMP      90               // OPCODE_START + OP_JMP
#define COL_BZ       92               // OPCODE_START + OP_BZ
#define COL_BNZ      93               // OPCODE_START + OP_BNZ

#define ROWS_PER_TILE 64
#define BLOCK         256
#define VEC_PER_TILE  (ROWS_PER_TILE * (ROW_F / 4))   // 3328 float4s
#define ISSUES        (VEC_PER_TILE / BLOCK)          // 13 (exact, no tail)
#define STRIDE_BYTES  (BLOCK * 16)                    // 4096 B per issue step

// ---------------------------------------------------------------------------
// Pass 1: global any-reduction of the three opcode predicates.
// flags bit0 = any_jmp, bit1 = any_bz, bit2 = any_bnz.
// ---------------------------------------------------------------------------
__global__ void cf_scan_kernel(const float* __restrict__ x,
                               unsigned* __restrict__ flags, int rows) {
    __shared__ unsigned sb;
    if (threadIdx.x == 0) sb = 0u;
    __syncthreads();

    int r = blockIdx.x * blockDim.x + threadIdx.x;
    unsigned b = 0u;
    if (r < rows) {
        const float* row = x + (long long)r * ROW_F;
        if (row[COL_JMP] > 0.5f) b |= 1u;
        if (row[COL_BZ]  > 0.5f) b |= 2u;
        if (row[COL_BNZ] > 0.5f) b |= 4u;
    }
    if (b) atomicOr(&sb, b);
    __syncthreads();
    if (threadIdx.x == 0 && sb) atomicOr(flags, sb);
}

// ---------------------------------------------------------------------------
// Pass 2: stream 64-row tiles memory -> LDS -> memory with CDNA5 async DMA
// (ASYNCcnt-tracked, non-temporal), patching PC nibbles + BRANCH_TAKEN in LDS.
// ---------------------------------------------------------------------------
__global__ void cf_rewrite_kernel(const float* __restrict__ x,
                                  float* __restrict__ out,
                                  const unsigned* __restrict__ flags,
                                  int rows) {
    __shared__ float tile[ROWS_PER_TILE * ROW_F];   // 53,248 bytes

    const long long rowBase = (long long)blockIdx.x * ROWS_PER_TILE;
    int tileRows = rows - (int)rowBase;
    if (tileRows <= 0) return;
    if (tileRows > ROWS_PER_TILE) tileRows = ROWS_PER_TILE;

    const float* __restrict__ srcF = x   + rowBase * ROW_F;
    float*       __restrict__ dstF = out + rowBase * ROW_F;

    // Hoist the uniform flag load so its latency overlaps the DMA burst.
    const unsigned f = *flags;

    // LDS byte address = low 32 bits of the generic shared-aperture pointer.
    const unsigned ldsBase = (unsigned)(size_t)(&tile[0]);
    const unsigned la0 = ldsBase + (unsigned)threadIdx.x * 16u;
    const unsigned long long ga0 =
        (unsigned long long)(const void*)srcF + (unsigned long long)threadIdx.x * 16ull;

    const bool fullTile = (tileRows == ROWS_PER_TILE);

    // ---- async load burst: memory -> LDS (read-once -> non-temporal) ----
    if (fullTile) {
        unsigned la = la0;
        unsigned long long ga = ga0;
#pragma unroll
        for (int k = 0; k < ISSUES; ++k) {
            asm volatile("global_load_async_to_lds_b128 %0, %1, off th:TH_LOAD_NT"
                         :: "v"(la), "v"(ga) : "memory");
            la += STRIDE_BYTES;
            ga += STRIDE_BYTES;
        }
    } else {
        const int nvec = tileRows * (ROW_F / 4);
        for (int i = threadIdx.x; i < nvec; i += BLOCK) {
            unsigned           la = ldsBase + (unsigned)i * 16u;
            unsigned long long ga =
                (unsigned long long)(const void*)srcF + (unsigned long long)i * 16ull;
            asm volatile("global_load_async_to_lds_b128 %0, %1, off th:TH_LOAD_NT"
                         :: "v"(la), "v"(ga) : "memory");
        }
    }
    asm volatile("s_wait_asynccnt 0" ::: "memory");
    __syncthreads();

    // ---- per-row patch in LDS (uniform branch: f identical grid-wide) ----
    if (f != 0u) {
        const bool fj = (f & 1u) != 0u;
        const bool fz = (f & 2u) != 0u;
        const bool fn = (f & 4u) != 0u;
        for (int r = threadIdx.x; r < tileRows; r += BLOCK) {
            float* row = &tile[r * ROW_F];

            int   ax  = 0;
            float pc  = 0.0f, immv = 0.0f;
            int   i16 = 1;
            float p16 = 1.0f;
#pragma unroll
            for (int i = 0; i < 8; ++i) {
                ax   += (int)row[AX_START  + i] * i16;   // astype(int32): trunc
                pc   += row[PC_START  + i] * p16;
                immv += row[IMM_START + i] * p16;
                i16 *= 16;
                p16 *= 16.0f;
            }
            const bool az = (ax == 0);

            float new_pc = fj ? immv
                         : fz ? (az ? immv : pc + 8.0f)
                         : fn ? (!az ? immv : pc + 8.0f)
                         : pc;
            float bt = fj ? 1.0f
                     : fz ? (az ? 1.0f : 0.0f)
                     : fn ? (az ? 0.0f : 1.0f)
                     : row[BRANCH_TAKEN];

            const int np = (int)new_pc;                 // trunc toward zero
#pragma unroll
            for (int i = 0; i < 8; ++i)                 // arithmetic >> == i32 >>
                row[PC_START + i] = (float)((np >> (4 * i)) & 15);
            row[BRANCH_TAKEN] = bt;
        }
        __syncthreads();
    }

    // ---- async store burst: LDS -> memory (write-once -> non-temporal) ----
    const unsigned long long gs0 =
        (unsigned long long)(void*)dstF + (unsigned long long)threadIdx.x * 16ull;
    if (fullTile) {
        unsigned la = la0;
        unsigned long long ga = gs0;
#pragma unroll
        for (int k = 0; k < ISSUES; ++k) {
            asm volatile("global_store_async_from_lds_b128 %0, %1, off th:TH_STORE_NT"
                         :: "v"(ga), "v"(la) : "memory");
            la += STRIDE_BYTES;
            ga += STRIDE_BYTES;
        }
    } else {
        const int nvec = tileRows * (ROW_F / 4);
        for (int i = threadIdx.x; i < nvec; i += BLOCK) {
            unsigned           la = ldsBase + (unsigned)i * 16u;
            unsigned long long ga =
                (unsigned long long)(void*)dstF + (unsigned long long)i * 16ull;
            asm volatile("global_store_async_from_lds_b128 %0, %1, off th:TH_STORE_NT"
                         :: "v"(ga), "v"(la) : "memory");
        }
    }
    asm volatile("s_wait_asynccnt 0" ::: "memory");
}

// ---------------------------------------------------------------------------
// Launch
// ---------------------------------------------------------------------------
extern "C" void kernel_launch(void* const* d_in, const int* in_sizes, int n_in,
                              void* d_out, int out_size, void* d_ws, size_t ws_size,
                              hipStream_t stream) {
    const float* x   = (const float*)d_in[0];
    float*       out = (float*)d_out;
    const int n    = in_sizes[0];          // 16*8192*208
    const int rows = n / ROW_F;            // 131072
    unsigned* flags = (unsigned*)d_ws;

    hipMemsetAsync(d_ws, 0, sizeof(unsigned), stream);   // capture-safe memset node

    const int scanBlocks = (rows + BLOCK - 1) / BLOCK;
    cf_scan_kernel<<<scanBlocks, BLOCK, 0, stream>>>(x, flags, rows);

    const int tiles = (rows + ROWS_PER_TILE - 1) / ROWS_PER_TILE;
    cf_rewrite_kernel<<<tiles, BLOCK, 0, stream>>>(x, out, flags, rows);
}